// GATNet_1013612282046
// MI455X (gfx1250) — compile-verified
//
#include <hip/hip_runtime.h>
#include <hip/hip_bf16.h>
#include <math.h>

typedef float v2f __attribute__((ext_vector_type(2)));
typedef float v8f __attribute__((ext_vector_type(8)));

#define N_NODES 16384
#define N_EDGES 262144
#define ETOT    (N_EDGES + N_NODES)   // edges + self loops = 278528
#define NB      128
#define FEAT    78
#define HEADS   10
#define OUT_DIM 128
#define PROT_LEN 1000
#define CONV_T  121                   // 128 - 8 + 1
#define NEG_SLOPE 0.2f

__device__ inline void edge_nodes(const int* __restrict__ ei, int j, int& s, int& d) {
    if (j < N_EDGES) { s = ei[j]; d = ei[N_EDGES + j]; }
    else             { s = d = j - N_EDGES; }
}

// ---------------------------------------------------------------------------
// fp32 WMMA GEMM: C[M,N] = act(A[M,K] @ B[K,N] + bias), C stride ldc.
// One wave per 16x16 tile. No predication in the hot loop: out-of-range
// rows/cols are index-clamped (their results are discarded at store time);
// only the K tail needs exact zeros, handled once after the loop.
// ---------------------------------------------------------------------------
__global__ void wmma_gemm_f32(const float* __restrict__ A, const float* __restrict__ B,
                              float* __restrict__ C, int M, int N, int K, int ldc,
                              const float* __restrict__ bias, int act) {
    const int lane = threadIdx.x & 31;
    const int wid  = threadIdx.x >> 5;
    const int tiles_n = (N + 15) >> 4;
    const int tiles_m = (M + 15) >> 4;
    const int tile = blockIdx.x * (blockDim.x >> 5) + wid;
    if (tile >= tiles_m * tiles_n) return;          // wave-uniform exit
    const int tm = tile / tiles_n, tn = tile % tiles_n;
    const int mrow = tm * 16 + (lane & 15);
    const int ncol = tn * 16 + (lane & 15);
    const int kh = (lane >> 4) << 1;                // 0 or 2 (K half select)
    const int arow = mrow < M ? mrow : (M - 1);     // clamp: loads stay in bounds
    const int bcol = ncol < N ? ncol : (N - 1);
    const int Kfull = K & ~3;
    const float* __restrict__ Ap = A + (size_t)arow * K + kh;
    const float* __restrict__ Bp = B + (size_t)kh * N + bcol;
    v8f acc = {};
#pragma unroll 2
    for (int k0 = 0; k0 < Kfull; k0 += 4) {
        v2f a, b;
        a.x = Ap[0]; a.y = Ap[1];                   // consecutive -> b64
        b.x = Bp[0]; b.y = Bp[N];
        acc = __builtin_amdgcn_wmma_f32_16x16x4_f32(false, a, false, b,
                                                    (short)0, acc, false, false);
        Ap += 4; Bp += (size_t)4 * N;
    }
    if (K & 3) {                                    // ragged K tail (once)
        const int ka = Kfull + kh;
        const int k0c = ka < K ? ka : (K - 1);
        const int k1c = ka + 1 < K ? ka + 1 : (K - 1);
        v2f a, b;
        a.x = A[(size_t)arow * K + k0c] * (ka     < K ? 1.f : 0.f);
        a.y = A[(size_t)arow * K + k1c] * (ka + 1 < K ? 1.f : 0.f);
        b.x = B[(size_t)k0c * N + bcol] * (ka     < K ? 1.f : 0.f);
        b.y = B[(size_t)k1c * N + bcol] * (ka + 1 < K ? 1.f : 0.f);
        acc = __builtin_amdgcn_wmma_f32_16x16x4_f32(false, a, false, b,
                                                    (short)0, acc, false, false);
    }
    if (ncol >= N) return;
    const float bi = bias ? bias[ncol] : 0.f;
    const int rbase = tm * 16 + ((lane >> 4) << 3);
#pragma unroll
    for (int i = 0; i < 8; ++i) {
        const int r = rbase + i;
        if (r < M) {
            float v = acc[i] + bi;
            if (act == 1) v = fmaxf(v, 0.f);
            C[(size_t)r * ldc + ncol] = v;
        }
    }
}

// ---------------------------------------------------------------------------
// Protein conv1d as gather-GEMM: M=128*121, K=8000, N=32 (exact 16-multiples).
// Loop over i (input channel = seq pos): one target gather per 8 K-values,
// two WMMAs per iteration, pairable b64 loads from emb and conv_w.
// ---------------------------------------------------------------------------
__global__ void prot_conv_gemm(const float* __restrict__ emb, const int* __restrict__ target,
                               const float* __restrict__ convw, const float* __restrict__ convb,
                               float* __restrict__ out) {
    const int lane = threadIdx.x & 31;
    const int wid  = threadIdx.x >> 5;
    const int tile = blockIdx.x * (blockDim.x >> 5) + wid;
    const int tiles_m = (NB * CONV_T) / 16;         // 968
    if (tile >= tiles_m * 2) return;
    const int tm = tile >> 1, tn = tile & 1;
    const int mrow = tm * 16 + (lane & 15);
    const int ocol = tn * 16 + (lane & 15);
    const int b = mrow / CONV_T, t = mrow % CONV_T;
    const int* __restrict__ trow = target + b * PROT_LEN;
    const int kh = (lane >> 4) << 1;
    const float* __restrict__ wrow = convw + (size_t)ocol * 8000 + kh;
    v8f acc = {};
    for (int i = 0; i < PROT_LEN; ++i) {
        const float* __restrict__ erow = emb + trow[i] * 128 + t + kh;
        v2f a0, b0, a1, b1;
        a0.x = erow[0]; a0.y = erow[1];
        b0.x = wrow[0]; b0.y = wrow[1];
        a1.x = erow[4]; a1.y = erow[5];
        b1.x = wrow[4]; b1.y = wrow[5];
        acc = __builtin_amdgcn_wmma_f32_16x16x4_f32(false, a0, false, b0,
                                                    (short)0, acc, false, false);
        acc = __builtin_amdgcn_wmma_f32_16x16x4_f32(false, a1, false, b1,
                                                    (short)0, acc, false, false);
        wrow += 8;
    }
    const float cb = convb[ocol];
    const int rbase = tm * 16 + ((lane >> 4) << 3);
#pragma unroll
    for (int i = 0; i < 8; ++i) {
        const int r = rbase + i;
        const int bb = r / CONV_T, tt = r % CONV_T;
        out[(size_t)bb * (32 * CONV_T) + ocol * CONV_T + tt] = fmaxf(acc[i] + cb, 0.f);
    }
}

// --------------------------- CSR build (deterministic) ---------------------
__global__ void fill_kernel(float* p, unsigned n, float v) {
    unsigned i = blockIdx.x * blockDim.x + threadIdx.x;
    if (i < n) p[i] = v;
}

__global__ void count_deg(const int* __restrict__ ei, int* __restrict__ deg) {
    int j = blockIdx.x * blockDim.x + threadIdx.x;
    if (j >= ETOT) return;
    int s, d; edge_nodes(ei, j, s, d);
    atomicAdd(&deg[d], 1);                          // integer: deterministic
}

__global__ void scan_rowptr(const int* __restrict__ deg, int* __restrict__ rowptr) {
    __shared__ int part[256];
    __shared__ int pref[257];
    const int t = threadIdx.x;
    const int base = t * 64;                        // 256 * 64 = 16384
    int s = 0;
    for (int i = 0; i < 64; ++i) s += deg[base + i];
    part[t] = s;
    __syncthreads();
    if (t == 0) {
        int run = 0;
        for (int i = 0; i < 256; ++i) { pref[i] = run; run += part[i]; }
        pref[256] = run;
    }
    __syncthreads();
    int run = pref[t];
    for (int i = 0; i < 64; ++i) { rowptr[base + i] = run; run += deg[base + i]; }
    if (t == 255) rowptr[N_NODES] = pref[256];
}

__global__ void copy_int(const int* __restrict__ a, int* __restrict__ b, int n) {
    int i = blockIdx.x * blockDim.x + threadIdx.x;
    if (i < n) b[i] = a[i];
}

__global__ void scatter_edges(const int* __restrict__ ei, int* __restrict__ cursor,
                              int* __restrict__ eidx) {
    int j = blockIdx.x * blockDim.x + threadIdx.x;
    if (j >= ETOT) return;
    int s, d; edge_nodes(ei, j, s, d);
    eidx[atomicAdd(&cursor[d], 1)] = j;             // slot order fixed by sort below
}

__global__ void sort_adj(const int* __restrict__ rowptr, int* __restrict__ eidx) {
    int n = blockIdx.x * blockDim.x + threadIdx.x;
    if (n >= N_NODES) return;
    const int lo = rowptr[n], hi = rowptr[n + 1];
    for (int i = lo + 1; i < hi; ++i) {             // deg ~ Poisson(17): tiny sorts
        int v = eidx[i], k = i - 1;
        while (k >= lo && eidx[k] > v) { eidx[k + 1] = eidx[k]; --k; }
        eidx[k + 1] = v;
    }
}

// --------------------------- attention ------------------------------------
// alpha[n,h] = sum_c xl[n,h,c] * att[h,c]   (src and dst in one pass)
__global__ void node_alpha(const float* __restrict__ xl, const float* __restrict__ wsrc,
                           const float* __restrict__ wdst, int H, int C,
                           float* __restrict__ as, float* __restrict__ ad) {
    int idx = blockIdx.x * blockDim.x + threadIdx.x;
    if (idx >= N_NODES * H) return;
    const int n = idx / H, h = idx % H;
    const float* row = xl + (size_t)n * H * C + h * C;
    const float* w1 = wsrc + h * C;
    const float* w2 = wdst + h * C;
    float s1 = 0.f, s2 = 0.f;
    for (int c = 0; c < C; ++c) { float v = row[c]; s1 += v * w1[c]; s2 += v * w2[c]; }
    as[idx] = s1; ad[idx] = s2;
}

// ---------------------------------------------------------------------------
// Gather-style GAT aggregation, one block per destination node.
// Deterministic (sorted edge list, fixed summation order), no float atomics.
// out[n,h,c] = act( sum_e softmax_coef(e,h) * xl[src_e,h,c] + bias[h,c] )
// act: 1=ReLU, 2=ELU
// ---------------------------------------------------------------------------
#define CHUNK 32
__global__ void gat_aggregate(const float* __restrict__ xl, const float* __restrict__ as,
                              const float* __restrict__ ad, const int* __restrict__ rowptr,
                              const int* __restrict__ eidx, const int* __restrict__ ei,
                              const float* __restrict__ bias, float* __restrict__ out,
                              int H, int C, int act) {
    const int n = blockIdx.x;
    const int t = threadIdx.x;
    const int bs = blockDim.x;
    const int lo = rowptr[n];
    const int deg = rowptr[n + 1] - lo;
    const int HC = H * C;
    __shared__ float sh_ad[16], sh_m[16], sh_inv[16];
    __shared__ int   sh_src[CHUNK];
    __shared__ float sh_coef[CHUNK * 16];
    if (t < H) sh_ad[t] = ad[n * H + t];
    __syncthreads();
    if (t < H) {                                    // per-head max then sum
        float m = -INFINITY;
        for (int q = 0; q < deg; ++q) {
            int j = eidx[lo + q], s, d; edge_nodes(ei, j, s, d);
            float v = as[s * H + t] + sh_ad[t];
            v = v > 0.f ? v : NEG_SLOPE * v;
            m = fmaxf(m, v);
        }
        float ssum = 0.f;
        for (int q = 0; q < deg; ++q) {
            int j = eidx[lo + q], s, d; edge_nodes(ei, j, s, d);
            float v = as[s * H + t] + sh_ad[t];
            v = v > 0.f ? v : NEG_SLOPE * v;
            ssum += expf(v - m);
        }
        sh_m[t] = m;
        sh_inv[t] = 1.f / (ssum + 1e-16f);
    }
    __syncthreads();
    // per-thread channel assignment (<= 4 channels/thread)
    int ccs[4], hh[4];
    float acc[4];
#pragma unroll
    for (int i = 0; i < 4; ++i) {
        ccs[i] = t + i * bs;
        hh[i] = ccs[i] < HC ? ccs[i] / C : 0;
        acc[i] = 0.f;
    }
    for (int q0 = 0; q0 < deg; q0 += CHUNK) {
        const int cn = min(CHUNK, deg - q0);
        if (t < cn) {
            int j = eidx[lo + q0 + t], s, d; edge_nodes(ei, j, s, d);
            sh_src[t] = s;
        }
        __syncthreads();
        for (int w = t; w < cn * H; w += bs) {
            const int e = w / H, h = w - e * H;
            float v = as[sh_src[e] * H + h] + sh_ad[h];
            v = v > 0.f ? v : NEG_SLOPE * v;
            sh_coef[e * H + h] = expf(v - sh_m[h]) * sh_inv[h];
        }
        __syncthreads();
        for (int e = 0; e < cn; ++e) {
            const float* __restrict__ row = xl + (size_t)sh_src[e] * HC;
#pragma unroll
            for (int i = 0; i < 4; ++i)
                if (ccs[i] < HC) acc[i] += sh_coef[e * H + hh[i]] * row[ccs[i]];
        }
        __syncthreads();
    }
#pragma unroll
    for (int i = 0; i < 4; ++i) {
        if (ccs[i] < HC) {
            float v = acc[i] + bias[ccs[i]];
            v = (act == 1) ? fmaxf(v, 0.f) : (v > 0.f ? v : (expf(v) - 1.f));
            out[(size_t)n * HC + ccs[i]] = v;
        }
    }
}

// global max pool (inputs >= 0 post-ReLU; max is order-independent)
__global__ void pool_max(const float* __restrict__ h, const int* __restrict__ batch,
                         float* __restrict__ g) {
    unsigned idx = blockIdx.x * blockDim.x + threadIdx.x;
    if (idx >= (unsigned)N_NODES * OUT_DIM) return;
    const int n = idx / OUT_DIM, c = idx % OUT_DIM;
    atomicMax((int*)&g[batch[n] * OUT_DIM + c], __float_as_int(h[idx]));
}

__global__ void final_out(const float* __restrict__ f2, const float* __restrict__ w,
                          const float* __restrict__ b, float* __restrict__ out, int K) {
    int r = blockIdx.x * blockDim.x + threadIdx.x;
    if (r >= NB) return;
    float acc = b[0];
    for (int k = 0; k < K; ++k) acc += f2[(size_t)r * K + k] * w[k];
    out[r] = acc;
}

// ---------------------------------------------------------------------------
extern "C" void kernel_launch(void* const* d_in, const int* in_sizes, int n_in,
                              void* d_out, int out_size, void* d_ws, size_t ws_size,
                              hipStream_t stream) {
    const float* x      = (const float*)d_in[0];
    const int*   ei     = (const int*)  d_in[1];
    const int*   batch  = (const int*)  d_in[2];
    const int*   target = (const int*)  d_in[3];
    const float* W1     = (const float*)d_in[4];
    const float* asrc1  = (const float*)d_in[5];
    const float* adst1  = (const float*)d_in[6];
    const float* b1     = (const float*)d_in[7];
    const float* W2     = (const float*)d_in[8];
    const float* asrc2  = (const float*)d_in[9];
    const float* adst2  = (const float*)d_in[10];
    const float* b2     = (const float*)d_in[11];
    const float* fcg1w  = (const float*)d_in[12];
    const float* fcg1b  = (const float*)d_in[13];
    const float* emb    = (const float*)d_in[14];
    const float* convw  = (const float*)d_in[15];
    const float* convb  = (const float*)d_in[16];
    const float* fcxtw  = (const float*)d_in[17];
    const float* fcxtb  = (const float*)d_in[18];
    const float* fc1w   = (const float*)d_in[19];
    const float* fc1b   = (const float*)d_in[20];
    const float* fc2w   = (const float*)d_in[21];
    const float* fc2b   = (const float*)d_in[22];
    const float* outw   = (const float*)d_in[23];
    const float* outb   = (const float*)d_in[24];
    float* out = (float*)d_out;

    // workspace carve-up (4-byte units)
    float* w = (float*)d_ws;
    size_t off = 0;
    auto alloc = [&](size_t n) { float* p = w + off; off += n; return p; };
    float* xl   = alloc((size_t)N_NODES * HEADS * FEAT);
    float* as1  = alloc((size_t)N_NODES * HEADS);
    float* ad1  = alloc((size_t)N_NODES * HEADS);
    float* h1   = alloc((size_t)N_NODES * HEADS * FEAT);
    float* h2l  = alloc((size_t)N_NODES * OUT_DIM);
    float* as2  = alloc(N_NODES);
    float* ad2  = alloc(N_NODES);
    float* h2   = alloc((size_t)N_NODES * OUT_DIM);
    float* g    = alloc((size_t)NB * OUT_DIM);
    float* xc   = alloc((size_t)NB * 256);
    float* conv = alloc((size_t)NB * 32 * CONV_T);
    float* f1   = alloc((size_t)NB * 1024);
    float* f2   = alloc((size_t)NB * 256);
    int* deg    = (int*)alloc(N_NODES);
    int* rowptr = (int*)alloc(N_NODES + 1);
    int* cursor = (int*)alloc(N_NODES);
    int* eidx   = (int*)alloc(ETOT);

    auto gemm = [&](const float* A, const float* B, float* C, int M, int N, int K,
                    int ldc, const float* bias, int act) {
        int tiles = ((M + 15) / 16) * ((N + 15) / 16);
        wmma_gemm_f32<<<(tiles + 3) / 4, 128, 0, stream>>>(A, B, C, M, N, K, ldc, bias, act);
    };

    // ---- CSR build (shared by both GAT layers; deterministic) ----
    fill_kernel<<<(N_NODES + 255) / 256, 256, 0, stream>>>((float*)deg, N_NODES, 0.f);
    count_deg<<<(ETOT + 255) / 256, 256, 0, stream>>>(ei, deg);
    scan_rowptr<<<1, 256, 0, stream>>>(deg, rowptr);
    copy_int<<<(N_NODES + 255) / 256, 256, 0, stream>>>(rowptr, cursor, N_NODES);
    scatter_edges<<<(ETOT + 255) / 256, 256, 0, stream>>>(ei, cursor, eidx);
    sort_adj<<<(N_NODES + 255) / 256, 256, 0, stream>>>(rowptr, eidx);

    // ---- GAT layer 1 (78 -> 10x78) ----
    gemm(x, W1, xl, N_NODES, HEADS * FEAT, FEAT, HEADS * FEAT, nullptr, 0);
    node_alpha<<<(N_NODES * HEADS + 255) / 256, 256, 0, stream>>>(xl, asrc1, adst1,
                                                                  HEADS, FEAT, as1, ad1);
    gat_aggregate<<<N_NODES, 256, 0, stream>>>(xl, as1, ad1, rowptr, eidx, ei,
                                               b1, h1, HEADS, FEAT, /*elu*/2);

    // ---- GAT layer 2 (780 -> 128) ----
    gemm(h1, W2, h2l, N_NODES, OUT_DIM, HEADS * FEAT, OUT_DIM, nullptr, 0);
    node_alpha<<<(N_NODES + 255) / 256, 256, 0, stream>>>(h2l, asrc2, adst2,
                                                          1, OUT_DIM, as2, ad2);
    gat_aggregate<<<N_NODES, 128, 0, stream>>>(h2l, as2, ad2, rowptr, eidx, ei,
                                               b2, h2, 1, OUT_DIM, /*relu*/1);

    // ---- pool + fc_g1 (writes concat cols 0..127) ----
    fill_kernel<<<((unsigned)NB * OUT_DIM + 255) / 256, 256, 0, stream>>>(g, NB * OUT_DIM, 0.f);
    pool_max<<<((unsigned)N_NODES * OUT_DIM + 255) / 256, 256, 0, stream>>>(h2, batch, g);
    gemm(g, fcg1w, xc, NB, OUT_DIM, OUT_DIM, 256, fcg1b, /*relu*/1);

    // ---- protein branch (writes concat cols 128..255) ----
    prot_conv_gemm<<<(968 * 2 + 3) / 4, 128, 0, stream>>>(emb, target, convw, convb, conv);
    gemm(conv, fcxtw, xc + OUT_DIM, NB, OUT_DIM, 32 * CONV_T, 256, fcxtb, 0);

    // ---- fused MLP head ----
    gemm(xc, fc1w, f1, NB, 1024, 256, 1024, fc1b, /*relu*/1);
    gemm(f1, fc2w, f2, NB, 256, 1024, 256, fc2b, /*relu*/1);
    final_out<<<1, NB, 0, stream>>>(f2, outw, outb, out, 256);
}